// SequentialSAGEConv_21689584844955
// MI455X (gfx1250) — compile-verified
//
#include <hip/hip_runtime.h>

#define N_NODES 50000
#define N_EDGES 800000

typedef __attribute__((ext_vector_type(2))) float v2f;
typedef __attribute__((ext_vector_type(8))) float v8f;

// ---------------- utility kernels ----------------

__global__ void zero_f32(float* __restrict__ p, long n) {
  long i = (long)blockIdx.x * blockDim.x + threadIdx.x;
  long stride = (long)gridDim.x * blockDim.x;
  for (; i < n; i += stride) p[i] = 0.0f;
}

__global__ void count_edges(const int* __restrict__ dst, float* __restrict__ cnt, int nE) {
  int e = blockIdx.x * blockDim.x + threadIdx.x;
  if (e < nE) atomicAdd(&cnt[dst[e]], 1.0f);
}

__global__ void recip_clamp(const float* __restrict__ cnt, float* __restrict__ inv, int n) {
  int i = blockIdx.x * blockDim.x + threadIdx.x;
  if (i < n) inv[i] = 1.0f / fmaxf(cnt[i], 1.0f);
}

// one wave (32 lanes) per edge; each lane scatter-adds d/32 feature columns.
// Gathers + atomics stay resident in MI455X's 192MB L2 (feature matrix <= 26MB).
__global__ void scatter_edges(const float* __restrict__ feat,
                              const int* __restrict__ src,
                              const int* __restrict__ dst,
                              float* __restrict__ agg,
                              int nE, int d) {
  int wave = blockIdx.x * (blockDim.x >> 5) + (threadIdx.x >> 5);
  int lane = threadIdx.x & 31;
  if (wave >= nE) return;
  const float* fs = feat + (size_t)src[wave] * d;
  float*       fd = agg  + (size_t)dst[wave] * d;
  for (int c = lane; c < d; c += 32)
    atomicAdd(&fd[c], fs[c]);
}

// ---------- fused mean-scale + dual GEMM via V_WMMA_F32_16X16X4_F32 --------
//
// out[m][j] = sum_k (agg[m][k]*inv[m]) * Wl[j][k]
//           + sum_k  xin[m][k]         * Wr[j][k]  + bias[j]
//
// One wave computes a 16(M) x 64(N) output block: 4 column tiles sharing the
// same A fragment per k-step (1 A load + 4 B loads feed 4 WMMAs).
// A (16x4 f32): lanes 0-15 -> K=0,1 ; lanes 16-31 -> K=2,3 (rows M=0..15)
// B (4x16 f32): symmetric (lane carries one column j, 2 consecutive K)
// C/D (16x16 f32): VGPR i: lanes 0-15 M=i, lanes 16-31 M=8+i; N=lane%16

template<int DIN, int DOUT>
__global__ __launch_bounds__(32)
void sage_wmma_gemm(const float* __restrict__ agg,
                    const float* __restrict__ xin,
                    const float* __restrict__ inv,
                    const float* __restrict__ Wl,   // [DOUT, DIN] row-major
                    const float* __restrict__ Wr,   // [DOUT, DIN] row-major
                    const float* __restrict__ bias, // [DOUT]
                    float* __restrict__ out) {      // [N_NODES, DOUT]
  const int lane = threadIdx.x & 31;
  const int half = lane >> 4;        // 0: lanes 0-15, 1: lanes 16-31
  const int l16  = lane & 15;
  const int mTile = blockIdx.x << 4; // node-row tile base
  const int nBase = blockIdx.y << 6; // 64-wide output-column block base

  const int aRow = mTile + l16;      // A row this lane feeds
  const int kOff = half << 1;        // K sub-offset (0 or 2)

  const float invc = inv[aRow];

  v8f acc[4] = {v8f{}, v8f{}, v8f{}, v8f{}};

  // mean-part: (agg * inv) @ Wl^T
  {
    const float* aP = agg + (size_t)aRow * DIN + kOff;
    const float* bP = Wl  + (size_t)(nBase + l16) * DIN + kOff;
#pragma unroll 4
    for (int k = 0; k < DIN; k += 4) {
      v2f a = *(const v2f*)(aP + k);
      a *= invc;
#pragma unroll
      for (int t = 0; t < 4; ++t) {
        v2f b = *(const v2f*)(bP + (size_t)t * 16 * DIN + k);
        acc[t] = __builtin_amdgcn_wmma_f32_16x16x4_f32(false, a, false, b,
                                                       (short)0, acc[t],
                                                       false, false);
      }
    }
  }

  // root-part: xin @ Wr^T
  {
    const float* aP = xin + (size_t)aRow * DIN + kOff;
    const float* bP = Wr  + (size_t)(nBase + l16) * DIN + kOff;
#pragma unroll 4
    for (int k = 0; k < DIN; k += 4) {
      v2f a = *(const v2f*)(aP + k);
#pragma unroll
      for (int t = 0; t < 4; ++t) {
        v2f b = *(const v2f*)(bP + (size_t)t * 16 * DIN + k);
        acc[t] = __builtin_amdgcn_wmma_f32_16x16x4_f32(false, a, false, b,
                                                       (short)0, acc[t],
                                                       false, false);
      }
    }
  }

  // writeback with bias
  const int rowBase = mTile + (half << 3);
#pragma unroll
  for (int t = 0; t < 4; ++t) {
    const int col = nBase + t * 16 + l16;
    const float bj = bias[col];
#pragma unroll
    for (int i = 0; i < 8; ++i)
      out[(size_t)(rowBase + i) * DOUT + col] = acc[t][i] + bj;
  }
}

// ---------------- host orchestration ----------------

extern "C" void kernel_launch(void* const* d_in, const int* in_sizes, int n_in,
                              void* d_out, int out_size, void* d_ws, size_t ws_size,
                              hipStream_t stream) {
  const float* x   = (const float*)d_in[0];
  const int*   ei  = (const int*)d_in[1];
  const int*   src = ei;
  const int*   dst = ei + N_EDGES;
  const float* Wl1 = (const float*)d_in[2];
  const float* b1  = (const float*)d_in[3];
  const float* Wr1 = (const float*)d_in[4];
  const float* Wl2 = (const float*)d_in[5];
  const float* b2  = (const float*)d_in[6];
  const float* Wr2 = (const float*)d_in[7];
  const float* Wl3 = (const float*)d_in[8];
  const float* b3  = (const float*)d_in[9];
  const float* Wr3 = (const float*)d_in[10];
  float* out = (float*)d_out;

  // workspace layout (floats)
  float* ws  = (float*)d_ws;
  float* cnt = ws;                                   //  50000 (+pad)
  float* inv = cnt + 50048;                          //  50000 (+pad)
  float* agg = inv + 50048;                          //  50000*128
  float* h1  = agg + (size_t)N_NODES * 128;          //  50000*128
  float* h2  = h1  + (size_t)N_NODES * 128;          //  50000*64
  // total ~64.4 MB

  // degree counts (graph is static -> once for all layers)
  zero_f32<<<256, 256, 0, stream>>>(cnt, (long)N_NODES);
  count_edges<<<(N_EDGES + 255) / 256, 256, 0, stream>>>(dst, cnt, N_EDGES);
  recip_clamp<<<(N_NODES + 255) / 256, 256, 0, stream>>>(cnt, inv, N_NODES);

  const int scatterBlocks = N_EDGES / 8;             // 8 waves of 32 per 256-thr block

  // ---- layer 1: 128 -> 128
  zero_f32<<<2048, 256, 0, stream>>>(agg, (long)N_NODES * 128);
  scatter_edges<<<scatterBlocks, 256, 0, stream>>>(x, src, dst, agg, N_EDGES, 128);
  {
    dim3 g(N_NODES / 16, 128 / 64);
    sage_wmma_gemm<128, 128><<<g, 32, 0, stream>>>(agg, x, inv, Wl1, Wr1, b1, h1);
  }

  // ---- layer 2: 128 -> 64
  zero_f32<<<2048, 256, 0, stream>>>(agg, (long)N_NODES * 128);
  scatter_edges<<<scatterBlocks, 256, 0, stream>>>(h1, src, dst, agg, N_EDGES, 128);
  {
    dim3 g(N_NODES / 16, 64 / 64);
    sage_wmma_gemm<128, 64><<<g, 32, 0, stream>>>(agg, h1, inv, Wl2, Wr2, b2, h2);
  }

  // ---- layer 3: 64 -> 128
  zero_f32<<<2048, 256, 0, stream>>>(agg, (long)N_NODES * 64);
  scatter_edges<<<scatterBlocks, 256, 0, stream>>>(h2, src, dst, agg, N_EDGES, 64);
  {
    dim3 g(N_NODES / 16, 128 / 64);
    sage_wmma_gemm<64, 128><<<g, 32, 0, stream>>>(agg, h2, inv, Wl3, Wr3, b3, out);
  }
}